// UniGCN_69561290326173
// MI455X (gfx1250) — compile-verified
//
#include <hip/hip_runtime.h>

typedef __attribute__((ext_vector_type(2))) float v2f;
typedef __attribute__((ext_vector_type(4))) float v4f;
typedef __attribute__((ext_vector_type(8))) float v8f;

// ---------------------------------------------------------------------------
// Kernel 1: per-graph tables + tiled target_rel output.
//   tgt_tab[b,d]  = x_input[rel_labels[b]+1, d]
//   head_tab[b,d] = node_feat[head_idx[b], d]
//   tail_tab[b,d] = node_feat[tail_idx[b], d]
//   out_target_rel[b, l*D+d] = tgt_tab[b,d]  for l in [0,L)
// ---------------------------------------------------------------------------
__global__ void unigcn_tables_kernel(const float* __restrict__ x_input,
                                     const float* __restrict__ node_feat,
                                     const int*   __restrict__ head_idx,
                                     const int*   __restrict__ tail_idx,
                                     const int*   __restrict__ rel_labels,
                                     float* __restrict__ out_target_rel,
                                     float* __restrict__ tgt_tab,
                                     float* __restrict__ head_tab,
                                     float* __restrict__ tail_tab,
                                     int B, int D, int L) {
    int gid = blockIdx.x * blockDim.x + threadIdx.x;
    if (gid >= B * D) return;
    int b = gid / D;
    int d = gid - b * D;
    float tr = x_input[(long)(rel_labels[b] + 1) * D + d];
    tgt_tab[gid]  = tr;
    head_tab[gid] = node_feat[(long)head_idx[b] * D + d];
    tail_tab[gid] = node_feat[(long)tail_idx[b] * D + d];
    for (int l = 0; l < L; ++l)
        out_target_rel[((long)b * L + l) * D + d] = tr;
}

// ---------------------------------------------------------------------------
// Kernel 2: fused edge-streaming gather. One thread per 128-bit chunk of a row
// (D=32 -> 8 threads per edge). All four [E,D] outputs are write-once, so use
// non-temporal stores to avoid polluting L2 with 512 MB of dead lines.
// ---------------------------------------------------------------------------
__global__ void unigcn_edge_kernel(const float* __restrict__ x_input,
                                   const int*   __restrict__ edge_type,
                                   const int*   __restrict__ edge_seg_id,
                                   const float* __restrict__ tgt_tab,
                                   const float* __restrict__ head_tab,
                                   const float* __restrict__ tail_tab,
                                   float* __restrict__ out_rel,
                                   float* __restrict__ out_tgt,
                                   float* __restrict__ out_head,
                                   float* __restrict__ out_tail,
                                   int E, int D) {
    const int chunks = D >> 2;                       // v4f chunks per row (8)
    long gid = (long)blockIdx.x * blockDim.x + threadIdx.x;
    long total = (long)E * chunks;
    if (gid >= total) return;
    int e    = (int)(gid / chunks);
    int part = (int)(gid - (long)e * chunks);

    int et  = edge_type[e] + 1;                      // index_offset = 1
    int seg = edge_seg_id[e];

    const v4f* rel_src  = (const v4f*)(x_input  + (long)et  * D);
    const v4f* tgt_src  = (const v4f*)(tgt_tab  + (long)seg * D);
    const v4f* head_src = (const v4f*)(head_tab + (long)seg * D);
    const v4f* tail_src = (const v4f*)(tail_tab + (long)seg * D);

    long o = (long)e * D;
    __builtin_nontemporal_store(rel_src[part],  ((v4f*)(out_rel  + o)) + part);
    __builtin_nontemporal_store(tgt_src[part],  ((v4f*)(out_tgt  + o)) + part);
    __builtin_nontemporal_store(head_src[part], ((v4f*)(out_head + o)) + part);
    __builtin_nontemporal_store(tail_src[part], ((v4f*)(out_tail + o)) + part);
}

// ---------------------------------------------------------------------------
// Kernel 3: g_feats = node_feat @ W1 + b1 via V_WMMA_F32_16X16X4_F32 (exact
// fp32, so numerics match the reference). One wave computes a 16x32 tile:
// two 16x16 accumulators, K=32 consumed in 8 steps of K=4.
//
// ISA fragment layouts (wave32):
//   A 16x4 f32 (2 VGPRs): lanes 0-15 hold {K=0,K=1} for row M=lane,
//                         lanes 16-31 hold {K=2,K=3} for row M=lane-16.
//   B 4x16 f32 (2 VGPRs): lanes 0-15 hold {K=0,K=1} for col N=lane,
//                         lanes 16-31 hold {K=2,K=3} for col N=lane-16.
//   C/D 16x16 f32 (8 VGPRs): VGPR r = row r (lanes 0-15) / row r+8 (16-31),
//                            col = lane & 15.
// Accumulators seeded with b1[col] -> bias add is free.
// ---------------------------------------------------------------------------
__global__ void unigcn_gemm_wmma_kernel(const float* __restrict__ A,   // [N,32]
                                        const float* __restrict__ W,   // [32,32]
                                        const float* __restrict__ bias,// [32]
                                        float* __restrict__ out,       // [N,32]
                                        int N) {
    const int D = 32;
    int lane = threadIdx.x & 31;
    int wave = threadIdx.x >> 5;
    int wavesPerBlock = blockDim.x >> 5;
    int row0 = (blockIdx.x * wavesPerBlock + wave) * 16;
    if (row0 >= N) return;                 // wave-uniform: EXEC stays all-ones

    int m     = lane & 15;                 // row-in-tile (A) / col-in-tile (B,C)
    int khalf = (lane >> 4) << 1;          // 0 for lanes 0-15, 2 for lanes 16-31

    float b0v = bias[m];
    float b1v = bias[16 + m];
    v8f c0, c1;
#pragma unroll
    for (int r = 0; r < 8; ++r) { c0[r] = b0v; c1[r] = b1v; }

    const float* arow = A + (long)(row0 + m) * D;

#pragma unroll
    for (int k0 = 0; k0 < D; k0 += 4) {
        v2f a, b0, b1;
        a.x  = arow[k0 + khalf];
        a.y  = arow[k0 + khalf + 1];
        b0.x = W[(k0 + khalf) * D + m];
        b0.y = W[(k0 + khalf + 1) * D + m];
        b1.x = W[(k0 + khalf) * D + 16 + m];
        b1.y = W[(k0 + khalf + 1) * D + 16 + m];
        c0 = __builtin_amdgcn_wmma_f32_16x16x4_f32(false, a, false, b0,
                                                   (short)0, c0, false, false);
        c1 = __builtin_amdgcn_wmma_f32_16x16x4_f32(false, a, false, b1,
                                                   (short)0, c1, false, false);
    }

    int mbase = row0 + ((lane >> 4) << 3); // lanes 16-31 own rows +8
#pragma unroll
    for (int r = 0; r < 8; ++r) {
        __builtin_nontemporal_store(c0[r], out + (long)(mbase + r) * D + m);
        __builtin_nontemporal_store(c1[r], out + (long)(mbase + r) * D + 16 + m);
    }
}

// ---------------------------------------------------------------------------
// Launcher
// ---------------------------------------------------------------------------
extern "C" void kernel_launch(void* const* d_in, const int* in_sizes, int n_in,
                              void* d_out, int out_size, void* d_ws, size_t ws_size,
                              hipStream_t stream) {
    const float* x_input   = (const float*)d_in[0];
    const float* node_feat = (const float*)d_in[1];
    const float* W1        = (const float*)d_in[2];
    const float* b1        = (const float*)d_in[3];
    const int* edge_type   = (const int*)d_in[4];
    const int* edge_seg_id = (const int*)d_in[5];
    const int* head_idx    = (const int*)d_in[6];
    const int* tail_idx    = (const int*)d_in[7];
    const int* rel_labels  = (const int*)d_in[8];
    // d_in[9] = num_layers (device scalar; L recovered from out_size instead)

    const int D = in_sizes[3];
    const int E = in_sizes[4];
    const int B = in_sizes[6];
    const int N = in_sizes[1] / D;

    const long edge_rows = (long)E * D;
    const long gfeat     = (long)N * D;
    const long tr_elems  = (long)out_size - 4 * edge_rows - gfeat;
    const int  L         = (int)(tr_elems / ((long)B * D));

    float* out_target_rel = (float*)d_out;              // [B, L*D]
    float* out_rel        = out_target_rel + tr_elems;  // [E, D]
    float* out_tgt        = out_rel  + edge_rows;       // [E, D]
    float* out_head       = out_tgt  + edge_rows;       // [E, D]
    float* out_tail       = out_head + edge_rows;       // [E, D]
    float* out_gfeats     = out_tail + edge_rows;       // [N, D]

    float* tgt_tab  = (float*)d_ws;                     // [B, D]
    float* head_tab = tgt_tab  + (long)B * D;
    float* tail_tab = head_tab + (long)B * D;

    // 1) per-graph tables + target_rel
    {
        int threads = 256;
        int blocks  = (B * D + threads - 1) / threads;
        unigcn_tables_kernel<<<blocks, threads, 0, stream>>>(
            x_input, node_feat, head_idx, tail_idx, rel_labels,
            out_target_rel, tgt_tab, head_tab, tail_tab, B, D, L);
    }

    // 2) fused edge-streaming gathers (4 outputs)
    {
        long total  = (long)E * (D >> 2);
        int threads = 256;
        long blocks = (total + threads - 1) / threads;
        unigcn_edge_kernel<<<(int)blocks, threads, 0, stream>>>(
            x_input, edge_type, edge_seg_id, tgt_tab, head_tab, tail_tab,
            out_rel, out_tgt, out_head, out_tail, E, D);
    }

    // 3) g_feats GEMM via fp32 WMMA
    {
        int threads = 256;                       // 8 waves/block
        int rowTiles = (N + 15) / 16;
        int blocks   = (rowTiles + 7) / 8;
        unigcn_gemm_wmma_kernel<<<blocks, threads, 0, stream>>>(
            node_feat, W1, b1, out_gfeats, N);
    }
}